// Decoder_62448824483930
// MI455X (gfx1250) — compile-verified
//
#include <hip/hip_runtime.h>
#include <math.h>

// Problem constants (from reference)
#define B_ 1024
#define E_ 128
#define H_ 256
#define S_ 100
#define T_ 100
#define CT_ 10.0f

typedef float v2f __attribute__((ext_vector_type(2)));
typedef float v8f __attribute__((ext_vector_type(8)));

__device__ __forceinline__ float wsum32(float v) {
#pragma unroll
  for (int o = 16; o; o >>= 1) v += __shfl_xor(v, o, 32);
  return v;
}
__device__ __forceinline__ float wmax32(float v) {
#pragma unroll
  for (int o = 16; o; o >>= 1) v = fmaxf(v, __shfl_xor(v, o, 32));
  return v;
}
__device__ __forceinline__ float sigmf(float x) { return 1.0f / (1.0f + expf(-x)); }

// ---------------------------------------------------------------------------
// Straight-line per-source WMMA K-loop. Loads index directly off the base
// pointers with 32-bit offsets so address-space inference keeps them GLOBAL
// (global_load_*), not flat (flat loads would tie up DScnt + LDS datapath).
// A-fragment (16x4 fp32): lane = m + 16*(k/2), reg = k%2.
// B-fragment (4x16):      lane = n + 16*(k/2), reg = k%2.
// ---------------------------------------------------------------------------
template <int WTRANS>
__device__ __forceinline__ void gemm_src(
    const float* __restrict__ A, int lda, int K,
    const float* __restrict__ W, int ldw,
    int row, int kh, int ml, int n0, v8f acc[4])
{
  int aoff = row * lda + 2 * kh;
  int woff[4];
#pragma unroll
  for (int j = 0; j < 4; ++j) {
    const int n = n0 + j * 16 + ml;
    woff[j] = WTRANS ? (n * ldw + 2 * kh) : ((2 * kh) * ldw + n);
  }
  const int wstep = WTRANS ? 4 : 4 * ldw;  // advance per k-step of 4
  const int wsec  = WTRANS ? 1 : ldw;      // offset of second k element
#pragma unroll 2
  for (int k = 0; k < K; k += 4) {
    v2f a; a.x = A[aoff]; a.y = A[aoff + 1];
    aoff += 4;
#pragma unroll
    for (int j = 0; j < 4; ++j) {
      v2f b; b.x = W[woff[j]]; b.y = W[woff[j] + wsec];
      woff[j] += wstep;
      acc[j] = __builtin_amdgcn_wmma_f32_16x16x4_f32(
          false, a, false, b, (short)0, acc[j], false, false);
    }
  }
}

// ---------------------------------------------------------------------------
// Generic fp32 WMMA GEMM:
//   C[m, n] = act( A1[m,:]@W1 (+ A2[m,:]@W2) + b1[n] (+ b2[n]) )
// Block: 256 threads = 8 waves; wave = 16(M) x 64(N) via 4 WMMA tiles;
// block tile 128x64. gridDim.z batches independent GEMMs (e_g/e_p over s).
// HAS2 selects the optional second source at compile time (no dead branches).
// ---------------------------------------------------------------------------
template <int WTRANS, int ACT, int HAS2>
__global__ __launch_bounds__(256) void gemm_k(
    const float* __restrict__ A1, int lda1, int K1, const float* __restrict__ W1, int ldw1,
    const float* __restrict__ A2, int lda2, int K2, const float* __restrict__ W2, int ldw2,
    const float* __restrict__ b1, const float* __restrict__ b2,
    float* __restrict__ C, int ldc,
    long strideAz, long strideCz)
{
  const int lane = threadIdx.x & 31;
  const int wv   = threadIdx.x >> 5;
  const int ml   = lane & 15;   // A: m / B,D: n
  const int kh   = lane >> 4;   // A,B: k-half ; D: m += 8*kh
  const int row  = blockIdx.x * 128 + wv * 16 + ml;
  const int n0   = blockIdx.y * 64;

  v8f acc[4] = {};
  gemm_src<WTRANS>(A1 + (long)blockIdx.z * strideAz, lda1, K1, W1, ldw1,
                   row, kh, ml, n0, acc);
  if (HAS2)
    gemm_src<WTRANS>(A2, lda2, K2, W2, ldw2, row, kh, ml, n0, acc);

  float* Cz = C + (long)blockIdx.z * strideCz;
  const int mhi = 8 * kh;
#pragma unroll
  for (int j = 0; j < 4; ++j) {
    const int n = n0 + j * 16 + ml;
    const float bv = (HAS2 ? (b1[n] + b2[n]) : b1[n]);
#pragma unroll
    for (int r = 0; r < 8; ++r) {
      float v = acc[j][r] + bv;
      if (ACT == 1) v = tanhf(v);
      const long mrow = blockIdx.x * 128 + wv * 16 + mhi + r;
      Cz[mrow * ldc + n] = v;
    }
  }
}

// ---------------------------------------------------------------------------
// LSTM elementwise: gates (B x 4H, order i,f,c,o) + cx -> cy, hy
// ---------------------------------------------------------------------------
__global__ __launch_bounds__(256) void lstm_k(
    const float* __restrict__ gates, const float* __restrict__ cx,
    float* __restrict__ cy, float* __restrict__ hy)
{
  const int id = blockIdx.x * 256 + threadIdx.x;  // 0 .. B*H-1
  const int b = id >> 8;                          // H_ == 256
  const int h = id & 255;
  const float* g = gates + (long)b * (4 * H_);
  const float ig = g[h], fg = g[H_ + h], cg = g[2 * H_ + h], og = g[3 * H_ + h];
  const float c = sigmf(fg) * cx[id] + sigmf(ig) * tanhf(cg);
  cy[id] = c;
  hy[id] = sigmf(og) * tanhf(c);
}

// LDS row stride for e-tiles: 101 dwords (odd -> conflict-free for both
// h-strided (pass1) and h-contiguous (pass2 / transpose-load) patterns).
#define ESTR 101

// ---------------------------------------------------------------------------
// Glimpse attention: one block per batch row b.
//   u[s] = sum_h gv[h] * tanh(gq[b,h] + e_g[b,s,h]); a = softmax_s(u)
//   g[b,h] = sum_s e_g[b,s,h] * a[s]
// e tile (~101KB) loaded to LDS once, used by both passes.
// ---------------------------------------------------------------------------
__global__ __launch_bounds__(256) void glimpse_k(
    const float* __restrict__ eg, const float* __restrict__ gq,
    const float* __restrict__ gv, float* __restrict__ gout)
{
  extern __shared__ float sm[];
  float* e_s = sm;                   // H_*ESTR
  float* q_s = e_s + H_ * ESTR;      // H_
  float* v_s = q_s + H_;             // H_
  float* u_s = v_s + H_;             // 128
  float* a_s = u_s + 128;            // 128

  const int b = blockIdx.x, tid = threadIdx.x;
  const float* ep = eg + (long)b * S_ * H_;
  for (int s = 0; s < S_; ++s) e_s[tid * ESTR + s] = ep[(long)s * H_ + tid]; // tid = h
  q_s[tid] = gq[(long)b * H_ + tid];
  v_s[tid] = gv[tid];
  __syncthreads();

  const int lane = tid & 31, wv = tid >> 5;
  for (int s = wv; s < S_; s += 8) {
    float p = 0.0f;
#pragma unroll
    for (int j = 0; j < 8; ++j) {
      const int h = lane + 32 * j;
      p += v_s[h] * tanhf(q_s[h] + e_s[h * ESTR + s]);
    }
    p = wsum32(p);
    if (lane == 0) u_s[s] = p;
  }
  __syncthreads();

  if (wv == 0) {  // softmax over s (one wave; lane handles s = lane + 32q)
    float mx = -3.4e38f;
#pragma unroll
    for (int q = 0; q < 4; ++q) { const int s = lane + 32 * q; if (s < S_) mx = fmaxf(mx, u_s[s]); }
    mx = wmax32(mx);
    float sum = 0.0f;
#pragma unroll
    for (int q = 0; q < 4; ++q) { const int s = lane + 32 * q;
      if (s < S_) { const float e = expf(u_s[s] - mx); a_s[s] = e; sum += e; } }
    sum = wsum32(sum);
    const float inv = 1.0f / sum;
#pragma unroll
    for (int q = 0; q < 4; ++q) { const int s = lane + 32 * q; if (s < S_) a_s[s] *= inv; }
  }
  __syncthreads();

  float acc = 0.0f;
  for (int s = 0; s < S_; ++s) acc += e_s[tid * ESTR + s] * a_s[s];
  gout[(long)b * H_ + tid] = acc;
}

// ---------------------------------------------------------------------------
// Pointer attention + masked argmax + mask update + gather of next input.
//   u[s] = sum_h pv[h]*tanh(pq[b,h]+e_p[b,s,h]); scores = softmax(10*tanh(u))
//   h_att = sum_s e_p * scores;  idx = first-argmax(mask? 0 : scores)
// ---------------------------------------------------------------------------
__global__ __launch_bounds__(256) void pointer_k(
    const float* __restrict__ epg, const float* __restrict__ pq,
    const float* __restrict__ pv, int* __restrict__ mask,
    const float* __restrict__ emb,
    float* __restrict__ scores_out, float* __restrict__ idx_out,
    float* __restrict__ hatt, float* __restrict__ dec_next)
{
  extern __shared__ float sm[];
  float* e_s  = sm;
  float* q_s  = e_s + H_ * ESTR;
  float* v_s  = q_s + H_;
  float* u_s  = v_s + H_;
  float* sc_s = u_s + 128;
  int*  idx_sh = (int*)(sc_s + 128);

  const int b = blockIdx.x, tid = threadIdx.x;
  const float* ep = epg + (long)b * S_ * H_;
  for (int s = 0; s < S_; ++s) e_s[tid * ESTR + s] = ep[(long)s * H_ + tid];
  q_s[tid] = pq[(long)b * H_ + tid];
  v_s[tid] = pv[tid];
  __syncthreads();

  const int lane = tid & 31, wv = tid >> 5;
  for (int s = wv; s < S_; s += 8) {
    float p = 0.0f;
#pragma unroll
    for (int j = 0; j < 8; ++j) {
      const int h = lane + 32 * j;
      p += v_s[h] * tanhf(q_s[h] + e_s[h * ESTR + s]);
    }
    p = wsum32(p);
    if (lane == 0) u_s[s] = p;
  }
  __syncthreads();

  if (wv == 0) {
    float lv[4];
    float mx = -3.4e38f;
#pragma unroll
    for (int q = 0; q < 4; ++q) { const int s = lane + 32 * q;
      if (s < S_) { lv[q] = CT_ * tanhf(u_s[s]); mx = fmaxf(mx, lv[q]); } }
    mx = wmax32(mx);
    float sum = 0.0f;
#pragma unroll
    for (int q = 0; q < 4; ++q) { const int s = lane + 32 * q;
      if (s < S_) { const float e = expf(lv[q] - mx); sc_s[s] = e; sum += e; } }
    sum = wsum32(sum);
    const float inv = 1.0f / sum;
#pragma unroll
    for (int q = 0; q < 4; ++q) { const int s = lane + 32 * q; if (s < S_) sc_s[s] *= inv; }
    // masked first-occurrence argmax (matches jnp.argmax tie-breaking)
    float bv = -1.0f; int bi = S_;
#pragma unroll
    for (int q = 0; q < 4; ++q) { const int s = lane + 32 * q;
      if (s < S_) { const float m = mask[(long)b * S_ + s] ? 0.0f : sc_s[s];
        if (m > bv) { bv = m; bi = s; } } }
#pragma unroll
    for (int o = 16; o; o >>= 1) {
      const float ov = __shfl_xor(bv, o, 32);
      const int   oi = __shfl_xor(bi, o, 32);
      if (ov > bv || (ov == bv && oi < bi)) { bv = ov; bi = oi; }
    }
    if (lane == 0) {
      idx_sh[0] = bi;
      mask[(long)b * S_ + bi] = 1;
      idx_out[b] = (float)bi;
    }
  }
  __syncthreads();

  if (tid < S_) scores_out[(long)b * S_ + tid] = sc_s[tid];
  const int ix = idx_sh[0];
  if (tid < E_) dec_next[(long)b * E_ + tid] = emb[((long)ix * B_ + b) * E_ + tid];

  float acc = 0.0f;
  for (int s = 0; s < S_; ++s) acc += e_s[tid * ESTR + s] * sc_s[s];
  hatt[(long)b * H_ + tid] = acc;
}

// ---------------------------------------------------------------------------
extern "C" void kernel_launch(void* const* d_in, const int* in_sizes, int n_in,
                              void* d_out, int out_size, void* d_ws, size_t ws_size,
                              hipStream_t stream)
{
  (void)in_sizes; (void)n_in; (void)out_size; (void)ws_size;
  const float* dec0 = (const float*)d_in[0];
  const float* emb  = (const float*)d_in[1];
  const float* hx0  = (const float*)d_in[2];
  const float* cx0  = (const float*)d_in[3];
  const float* ctx  = (const float*)d_in[4];
  const float* Wi = (const float*)d_in[5];  const float* bi  = (const float*)d_in[6];
  const float* Wh = (const float*)d_in[7];  const float* bh  = (const float*)d_in[8];
  const float* Wo = (const float*)d_in[9];  const float* bo  = (const float*)d_in[10];
  const float* Pq = (const float*)d_in[11]; const float* pbq = (const float*)d_in[12];
  const float* Pr = (const float*)d_in[13]; const float* pbr = (const float*)d_in[14];
  const float* pv = (const float*)d_in[15];
  const float* Gq = (const float*)d_in[16]; const float* gbq = (const float*)d_in[17];
  const float* Gr = (const float*)d_in[18]; const float* gbr = (const float*)d_in[19];
  const float* gv = (const float*)d_in[20];

  float* ws = (float*)d_ws;
  size_t o = 0;
  float* EG   = ws + o; o += (size_t)B_ * S_ * H_;   // e_g  [b][s][h]
  float* EP   = ws + o; o += (size_t)B_ * S_ * H_;   // e_p  [b][s][h]
  float* GT   = ws + o; o += (size_t)B_ * 4 * H_;    // gates
  float* HY   = ws + o; o += (size_t)B_ * H_;
  float* GQv  = ws + o; o += (size_t)B_ * H_;
  float* Gg   = ws + o; o += (size_t)B_ * H_;
  float* PQv  = ws + o; o += (size_t)B_ * H_;
  float* HATT = ws + o; o += (size_t)B_ * H_;
  float* HTL[2]; HTL[0] = ws + o; o += (size_t)B_ * H_; HTL[1] = ws + o; o += (size_t)B_ * H_;
  float* CY[2];  CY[0]  = ws + o; o += (size_t)B_ * H_; CY[1]  = ws + o; o += (size_t)B_ * H_;
  float* DEC[2]; DEC[0] = ws + o; o += (size_t)B_ * E_; DEC[1] = ws + o; o += (size_t)B_ * E_;
  int* MASK = (int*)(ws + o);

  float* out        = (float*)d_out;
  float* out_scores = out;                                // (T,B,S)
  float* out_idx    = out + (size_t)T_ * B_ * S_;         // (T,B)
  float* out_hx     = out_idx + (size_t)T_ * B_;          // (B,H)
  float* out_cx     = out_hx + (size_t)B_ * H_;           // (B,H)

  hipMemsetAsync(MASK, 0, sizeof(int) * (size_t)B_ * S_, stream);

  const dim3 blk(256);
  // Precompute e_g, e_p once: for each s, C[b,h] = ctx_s[b,:] @ Gr^T + gbr  (WTRANS=1)
  gemm_k<1, 0, 0><<<dim3(B_ / 128, H_ / 64, S_), blk, 0, stream>>>(
      ctx, H_, H_, Gr, H_, nullptr, 0, 0, nullptr, 0,
      gbr, nullptr, EG, S_ * H_, (long)B_ * H_, (long)H_);
  gemm_k<1, 0, 0><<<dim3(B_ / 128, H_ / 64, S_), blk, 0, stream>>>(
      ctx, H_, H_, Pr, H_, nullptr, 0, 0, nullptr, 0,
      pbr, nullptr, EP, S_ * H_, (long)B_ * H_, (long)H_);

  const size_t smg = (size_t)(H_ * ESTR + H_ + H_ + 128 + 128) * sizeof(float);
  const size_t smp = smg + 16;

  for (int t = 0; t < T_; ++t) {
    const float* din = (t == 0) ? dec0 : DEC[(t - 1) & 1];
    const float* hin = (t == 0) ? hx0  : HTL[(t - 1) & 1];
    const float* cin = (t == 0) ? cx0  : CY[(t - 1) & 1];
    float* cyo = (t == T_ - 1) ? out_cx : CY[t & 1];
    float* hto = (t == T_ - 1) ? out_hx : HTL[t & 1];

    // gates = din @ Wi + hin @ Wh + (bi + bh)
    gemm_k<0, 0, 1><<<dim3(B_ / 128, 4 * H_ / 64, 1), blk, 0, stream>>>(
        din, E_, E_, Wi, 4 * H_, hin, H_, H_, Wh, 4 * H_,
        bi, bh, GT, 4 * H_, 0, 0);
    lstm_k<<<dim3((B_ * H_) / 256), blk, 0, stream>>>(GT, cin, cyo, HY);

    // gq = hy @ Gq + gbq
    gemm_k<0, 0, 0><<<dim3(B_ / 128, H_ / 64, 1), blk, 0, stream>>>(
        HY, H_, H_, Gq, H_, nullptr, 0, 0, nullptr, 0,
        gbq, nullptr, GQv, H_, 0, 0);
    glimpse_k<<<dim3(B_), blk, smg, stream>>>(EG, GQv, gv, Gg);

    // pq = g @ Pq + pbq
    gemm_k<0, 0, 0><<<dim3(B_ / 128, H_ / 64, 1), blk, 0, stream>>>(
        Gg, H_, H_, Pq, H_, nullptr, 0, 0, nullptr, 0,
        pbq, nullptr, PQv, H_, 0, 0);
    pointer_k<<<dim3(B_), blk, smp, stream>>>(
        EP, PQv, pv, MASK, emb,
        out_scores + (size_t)t * B_ * S_, out_idx + (size_t)t * B_,
        HATT, DEC[t & 1]);

    // h_tilde = tanh([h_att | hy] @ Wo + bo)   (Wo rows 0:H -> h_att, H:2H -> hy)
    gemm_k<0, 1, 1><<<dim3(B_ / 128, H_ / 64, 1), blk, 0, stream>>>(
        HATT, H_, H_, Wo, H_, HY, H_, H_, Wo + (size_t)H_ * H_, H_,
        bo, nullptr, hto, H_, 0, 0);
  }
}